// Bilinear_31379031065270
// MI455X (gfx1250) — compile-verified
//
#include <hip/hip_runtime.h>
#include <stdint.h>

// Problem dimensions (match the reference).
#define BB 128
#define HH 224
#define WW 224

struct __align__(4) F3 { float r, g, b; };

__device__ __forceinline__ F3 ld3(const float* __restrict__ p) {
  F3 v;
  __builtin_memcpy(&v, p, sizeof(F3));   // 12B @ 4B align -> global_load_b96
  return v;
}

// Cheap floor-mod into [0, m): v - m*floor(v/m), then integer clamp absorbs
// the rare rounding-edge (avoids the compiler's huge inline fmodf expansion).
__device__ __forceinline__ int wrap_idx(float v, float m, float inv_m, int mi) {
  float f = __builtin_fmaf(-m, floorf(v * inv_m), v);  // v - m*floor(v/m)
  int i = (int)f;
  i = i < 0 ? 0 : i;
  i = i > mi ? mi : i;
  return i;
}

// One block per (batch, row); one thread per output pixel along x.
__global__ __launch_bounds__(WW) void smooth_gather_kernel(
    const float* __restrict__ x, float* __restrict__ out) {
  const int y  = blockIdx.x;     // 0..223
  const int b  = blockIdx.y;     // 0..127
  const int tx = threadIdx.x;    // 0..223

  __shared__ float sD[2 * WW];   // staged (dx,dy) per pixel of this row

  const unsigned imgBase = (unsigned)b * (HH * WW * 5u);
  const unsigned pix     = imgBase + ((unsigned)y * WW + (unsigned)tx) * 5u;

  // ---- CDNA5 ASYNCcnt path: async global->LDS staging of (dx,dy) ----
  // Each lane copies its own 8-byte (dx,dy) record into its own LDS slot;
  // same-lane produce/consume, so s_wait_asynccnt 0 suffices (no barrier).
  {
    unsigned long long ga = (unsigned long long)(const void*)(x + pix + 3);
    unsigned lo = (unsigned)(uintptr_t)(const void*)&sD[2 * tx];
    asm volatile("global_load_async_to_lds_b64 %0, %1, off"
                 :: "v"(lo), "v"(ga)
                 : "memory");
    asm volatile("s_wait_asynccnt 0" ::: "memory");
  }
  const float dx = sD[2 * tx];
  const float dy = sD[2 * tx + 1];

  const int Xi = wrap_idx((float)tx + dx, 224.0f, 1.0f / 224.0f, WW - 1);
  const int Yi = wrap_idx((float)y  + dy, 224.0f, 1.0f / 224.0f, HH - 1);

  // sbp[Yi,Xi] reconstructed on the fly:
  //   border ring (pad) -> 0
  //   interior -> 0.25 * (img[Yi-1,Xi-1] + img[Yi-1,Xi+1] + img[Yi+1,Xi-1] + img[Yi+1,Xi+1])
  float r = 0.f, g = 0.f, bch = 0.f;
  if (Xi > 0 && Xi < WW - 1 && Yi > 0 && Yi < HH - 1) {
    const unsigned p0 = imgBase + ((unsigned)(Yi - 1) * WW + (unsigned)(Xi - 1)) * 5u;
    const F3 tl = ld3(x + p0);
    const F3 tr = ld3(x + p0 + 2u * 5u);
    const F3 bl = ld3(x + p0 + 2u * WW * 5u);
    const F3 br = ld3(x + p0 + 2u * WW * 5u + 2u * 5u);
    r   = 0.25f * ((tl.r + tr.r) + (bl.r + br.r));
    g   = 0.25f * ((tl.g + tr.g) + (bl.g + br.g));
    bch = 0.25f * ((tl.b + tr.b) + (bl.b + br.b));
  }

  // Non-temporal output stores: output is write-once (77 MB); keep the
  // 128 MB input resident in the 192 MB L2 for the gather re-reads.
  const unsigned o = (((unsigned)b * HH + (unsigned)y) * WW + (unsigned)tx) * 3u;
  __builtin_nontemporal_store(r,   out + o + 0);
  __builtin_nontemporal_store(g,   out + o + 1);
  __builtin_nontemporal_store(bch, out + o + 2);
}

extern "C" void kernel_launch(void* const* d_in, const int* in_sizes, int n_in,
                              void* d_out, int out_size, void* d_ws, size_t ws_size,
                              hipStream_t stream) {
  (void)in_sizes; (void)n_in; (void)out_size; (void)d_ws; (void)ws_size;
  const float* x = (const float*)d_in[0];
  float* out = (float*)d_out;

  dim3 grid(HH, BB);   // one block per (row, batch)
  dim3 block(WW);      // 224 threads = 7 wave32s
  hipLaunchKernelGGL(smooth_gather_kernel, grid, block, 0, stream, x, out);
}